// yolo_Loss_18219251269683
// MI455X (gfx1250) — compile-verified
//
#include <hip/hip_runtime.h>
#include <stdint.h>

// YOLO-v1 loss, forward only. Memory-bound: 193 MB in, 4 B out (~8.3us floor
// at 23.3 TB/s). No inner-product structure -> WMMA inapplicable; the CDNA5
// lever is the async global->LDS data mover (ASYNCcnt, b128 form) + wave32
// LDS compute + deterministic two-stage reduction.

namespace {
constexpr int   kS           = 7;
constexpr int   kN           = 30;                 // CLS(20) + B(2)*5
constexpr int   kBatch       = 16384;
constexpr int   kCells       = kBatch * kS * kS;   // 802816
constexpr int   kTPB         = 256;                // 8 wave32s
constexpr int   kBlocks      = kCells / kTPB;      // 3136 (exact fit)
constexpr float kLambdaCoord = 5.0f;
constexpr float kLambdaNoobj = 0.5f;

constexpr int kCellBytes    = kN * 4;               // 120
constexpr int kRegionBytes  = kTPB * kCellBytes;    // 30720 per tensor
constexpr int kRegionFloats = kTPB * kN;            // 7680 floats
constexpr int kChunkBytes   = kTPB * 16;            // 4096 per block-iteration
constexpr int kIters        = 2 * kRegionBytes / kChunkBytes;  // 15
}  // namespace

__device__ __forceinline__ float box_iou(float px, float py, float pw, float ph,
                                         float tx1, float ty1, float tx2, float ty2,
                                         float area2) {
  float x1 = px - pw * 0.5f, y1 = py - ph * 0.5f;
  float x2 = px + pw * 0.5f, y2 = py + ph * 0.5f;
  float ltx = fmaxf(x1, tx1), lty = fmaxf(y1, ty1);
  float rbx = fminf(x2, tx2), rby = fminf(y2, ty2);
  float w = fmaxf(rbx - ltx, 0.0f), h = fmaxf(rby - lty, 0.0f);
  float inter = w * h;
  float area1 = (x2 - x1) * (y2 - y1);
  float uni = area1 + area2 - inter;
  return uni > 0.0f ? inter / uni : 0.0f;
}

__global__ __launch_bounds__(kTPB) void yolo_loss_cells(
    const float* __restrict__ pred, const float* __restrict__ targ,
    float* __restrict__ partial) {
  // [pred: 30720 B][targ: 30720 B], cell c of this block at byte c*120.
  __shared__ __align__(16) float stage[2 * kRegionFloats];

  const int tid = (int)threadIdx.x;

  // Raw LDS byte offset: flat LDS addresses keep the aperture in the high
  // 32 bits and the LDS offset in addr[31:0], so truncation is exact.
  const uint32_t ldsBase = (uint32_t)(uintptr_t)(&stage[0]);
  // Byte offset of this block's 256-cell chunk in pred/targ.
  const uint32_t gBase = (uint32_t)blockIdx.x * (uint32_t)kRegionBytes;

  const unsigned long long predA = (unsigned long long)(uintptr_t)pred;
  const unsigned long long targA = (unsigned long long)(uintptr_t)targ;

  // Block-wide async staging: 15 x global_load_async_to_lds_b128 per thread,
  // 512 B per wave-instruction, fully coalesced, tracked by ASYNCcnt.
  // Iterations 0-6 are compile-time pred, 8-14 compile-time targ; i==7
  // straddles the 30720 B boundary but the branch is wave-uniform (boundary
  // is a multiple of the 512 B wave-chunk), so EXEC stays all-ones per op.
#pragma unroll
  for (int i = 0; i < kIters; ++i) {
    const uint32_t chunk = (uint32_t)(i * kChunkBytes) + (uint32_t)tid * 16u;
    if (chunk < (uint32_t)kRegionBytes) {
      asm volatile("global_load_async_to_lds_b128 %0, %1, %2"
                   :
                   : "v"(ldsBase + chunk), "v"(gBase + chunk), "s"(predA)
                   : "memory");
    } else {
      asm volatile("global_load_async_to_lds_b128 %0, %1, %2"
                   :
                   : "v"(ldsBase + chunk),
                     "v"(gBase + (chunk - (uint32_t)kRegionBytes)), "s"(targA)
                   : "memory");
    }
  }
  // Own async writes done...
  asm volatile("s_wait_asynccnt 0x0" ::: "memory");
  // ...and everyone else's (staging is block-interleaved).
  __syncthreads();

  // Compute out of LDS. Cell stride = 30 dwords; gcd(30,64)=2 gives 32
  // distinct banks across the wave -> conflict-free ds_load_b64.
  const float2* __restrict__ p2 =
      reinterpret_cast<const float2*>(&stage[tid * kN]);
  const float2* __restrict__ t2 =
      reinterpret_cast<const float2*>(&stage[kRegionFloats + tid * kN]);

  float pb[10], tb[10];
#pragma unroll
  for (int i = 0; i < 5; ++i) {
    float2 a = p2[i];
    pb[2 * i + 0] = a.x;
    pb[2 * i + 1] = a.y;
    float2 b = t2[i];
    tb[2 * i + 0] = b.x;
    tb[2 * i + 1] = b.y;
  }

  // class_loss: sum_{k=10..29}(p-t)^2 (obj mask applied at the end)
  float cls = 0.0f;
#pragma unroll
  for (int i = 5; i < 15; ++i) {
    float2 a = p2[i];
    float2 b = t2[i];
    float dx = a.x - b.x;
    float dy = a.y - b.y;
    cls += dx * dx + dy * dy;
  }

  const float t4 = tb[4];
  const float obj = (t4 == 1.0f) ? 1.0f : 0.0f;
  const float noobj = (t4 == 0.0f) ? 1.0f : 0.0f;

  // Reference quirk: conf_loc = arange(B)*B + 4 = {4, 6} (NOT {4, 9}).
  const float d4 = pb[4] - tb[4];
  const float d6 = pb[6] - tb[6];
  const float noobj_conf = d4 * d4 + d6 * d6;

  const float ttx1 = tb[0] - tb[2] * 0.5f;
  const float tty1 = tb[1] - tb[3] * 0.5f;
  const float ttx2 = tb[0] + tb[2] * 0.5f;
  const float tty2 = tb[1] + tb[3] * 0.5f;
  const float area2 = (ttx2 - ttx1) * (tty2 - tty1);

  const float iou0 =
      box_iou(pb[0], pb[1], pb[2], pb[3], ttx1, tty1, ttx2, tty2, area2);
  const float iou1 =
      box_iou(pb[5], pb[6], pb[7], pb[8], ttx1, tty1, ttx2, tty2, area2);

  // jnp.argmax: first occurrence wins on tie -> box 1 only if strictly greater.
  const bool j1 = iou1 > iou0;
  const float miou = fmaxf(iou0, iou1);

  const float rx = j1 ? pb[5] : pb[0];
  const float ry = j1 ? pb[6] : pb[1];
  const float rw = j1 ? pb[7] : pb[2];
  const float rh = j1 ? pb[8] : pb[3];
  const float rc = j1 ? pb[9] : pb[4];

  const float ex = rx - tb[0];
  const float ey = ry - tb[1];
  const float xy = ex * ex + ey * ey;

  const float sw = sqrtf(fmaxf(rw, 0.0f)) - sqrtf(fmaxf(tb[2], 0.0f));
  const float sh = sqrtf(fmaxf(rh, 0.0f)) - sqrtf(fmaxf(tb[3], 0.0f));
  const float wh = sw * sw + sh * sh;

  const float dc = rc - miou;
  const float obj_conf = dc * dc;

  const float cell_total =
      obj * (kLambdaCoord * (xy + wh) + obj_conf + cls) +
      kLambdaNoobj * noobj * noobj_conf;

  // ---- deterministic block reduction (LDS tree) ----
  __shared__ float red[kTPB];
  red[threadIdx.x] = cell_total;
  __syncthreads();
#pragma unroll
  for (int s = kTPB / 2; s > 0; s >>= 1) {
    if ((int)threadIdx.x < s) red[threadIdx.x] += red[threadIdx.x + s];
    __syncthreads();
  }
  if (threadIdx.x == 0) partial[blockIdx.x] = red[0];
}

__global__ __launch_bounds__(kTPB) void yolo_loss_final(
    const float* __restrict__ partial, float* __restrict__ out) {
  float s = 0.0f;
  for (int i = (int)threadIdx.x; i < kBlocks; i += kTPB) s += partial[i];

  __shared__ float red[kTPB];
  red[threadIdx.x] = s;
  __syncthreads();
#pragma unroll
  for (int k = kTPB / 2; k > 0; k >>= 1) {
    if ((int)threadIdx.x < k) red[threadIdx.x] += red[threadIdx.x + k];
    __syncthreads();
  }
  if (threadIdx.x == 0) out[0] = red[0] * (1.0f / (float)kBatch);
}

extern "C" void kernel_launch(void* const* d_in, const int* in_sizes, int n_in,
                              void* d_out, int out_size, void* d_ws, size_t ws_size,
                              hipStream_t stream) {
  (void)in_sizes;
  (void)n_in;
  (void)out_size;
  (void)ws_size;
  const float* pred = (const float*)d_in[0];
  const float* targ = (const float*)d_in[1];
  float* out = (float*)d_out;
  float* partials = (float*)d_ws;  // kBlocks floats, fully overwritten each call

  yolo_loss_cells<<<kBlocks, kTPB, 0, stream>>>(pred, targ, partials);
  yolo_loss_final<<<1, kTPB, 0, stream>>>(partials, out);
}